// SlayerVisMLP_59407987638665
// MI455X (gfx1250) — compile-verified
//
#include <hip/hip_runtime.h>

// ---------------------------------------------------------------------------
// SLAYER 3-layer SNN forward for MI455X (gfx1250, wave32).
//   stage1 : fused psp(x) -> spike -> psp(s1) => P1 [B,F,TPAD]   (2 series/thread)
//   fc1    : fp32 WMMA GEMM, split-K=7 -> U2p, then deterministic reduce -> U2
//   stage2 : spike(U2) -> psp => P2 [B,HID,TPAD]
//   fc2    : dense U3[b,q,t] = w2[q,:] . P2[b,:,t]   (fully parallel)
//   stage3 : final refractory spike over U3 -> out [B,OUT,T]
// ---------------------------------------------------------------------------

typedef float v2f __attribute__((ext_vector_type(2)));
typedef float v8f __attribute__((ext_vector_type(8)));

#define NB     8
#define F_IN   6300
#define TT     500
#define TPAD   512
#define HID    32
#define OUTN   20
#define K0N    (F_IN / 4)     // 1575 K-steps of 4
#define NTILES (TPAD / 16)    // 32 t-tiles
#define KSPLIT 7              // 1575 = 7 * 225
#define KCHUNK (K0N / KSPLIT) // 225
#define U2ELEMS ((size_t)NB * HID * TPAD)   // 131072

// SLAYER constants (double-accurate literals)
#define D_SR   0.77880078307140487f   // exp(-1/4)
#define A_SR   0.67957045711476130f   // e/4
#define D_REF  0.36787944117144233f   // exp(-1)
#define C_REF  -5.43656365691809047f  // -2e

// ---------------------------------------------------------------------------
// Pack w1 [32,6300] into WMMA A-fragment order (16x4 f32 tiles):
//   pack[((mtile*K0N + k0)*32 + lane)*2 + j] = w1[mtile*16 + (lane&15)]
//                                               [4*k0 + 2*(lane>>4) + j]
// ---------------------------------------------------------------------------
__global__ void repack_w1_kernel(const float* __restrict__ w1,
                                 float* __restrict__ pack) {
    int idx = blockIdx.x * blockDim.x + threadIdx.x;
    const int TOT = 2 * K0N * 64;
    if (idx >= TOT) return;
    int mtile = idx / (K0N * 64);
    int r     = idx % (K0N * 64);
    int k0    = r / 64;
    int e     = r % 64;
    int lane  = e >> 1;
    int j     = e & 1;
    int m = mtile * 16 + (lane & 15);
    int k = k0 * 4 + ((lane >> 4) << 1) + j;
    pack[idx] = w1[m * F_IN + k];
}

// ---------------------------------------------------------------------------
// Stage 1: fused psp -> spike -> psp. Two independent series per thread for
// ILP (dependent ~8-op VALU chain per timestep is latency-exposed at low
// waves/SIMD). float4 streaming loads/stores; zero-fill T padding.
// ---------------------------------------------------------------------------
#define NSER   (NB * F_IN)        // 50400
#define HSER   (NSER / 2)         // 25200

__global__ void stage1_kernel(const float* __restrict__ x,
                              float* __restrict__ P1) {
    int idx = blockIdx.x * blockDim.x + threadIdx.x;
    if (idx >= HSER) return;
    const float4* xin0  = (const float4*)(x  + (size_t)idx * TT);
    const float4* xin1  = (const float4*)(x  + (size_t)(idx + HSER) * TT);
    float4*       pout0 = (float4*)(P1 + (size_t)idx * TPAD);
    float4*       pout1 = (float4*)(P1 + (size_t)(idx + HSER) * TPAD);

    float pz0=0.f, py0=0.f, rz0=0.f, ry0=0.f, qz0=0.f, qy0=0.f;
    float pz1=0.f, py1=0.f, rz1=0.f, ry1=0.f, qz1=0.f, qy1=0.f;

    for (int t4 = 0; t4 < TT / 4; ++t4) {
        float4 xv0 = xin0[t4];
        float4 xv1 = xin1[t4];
        float xs0[4] = {xv0.x, xv0.y, xv0.z, xv0.w};
        float xs1[4] = {xv1.x, xv1.y, xv1.z, xv1.w};
        float os0[4], os1[4];
#pragma unroll
        for (int j = 0; j < 4; ++j) {
            // series 0
            py0 = D_SR * (py0 + pz0);  pz0 = D_SR * pz0 + xs0[j];
            float p0 = A_SR * py0;
            ry0 = D_REF * (ry0 + rz0);
            float u0 = p0 + C_REF * ry0;
            float s0 = (u0 >= 1.0f) ? 1.0f : 0.0f;
            rz0 = D_REF * rz0 + s0;
            qy0 = D_SR * (qy0 + qz0);  qz0 = D_SR * qz0 + s0;
            os0[j] = A_SR * qy0;
            // series 1 (independent chain -> ILP)
            py1 = D_SR * (py1 + pz1);  pz1 = D_SR * pz1 + xs1[j];
            float p1 = A_SR * py1;
            ry1 = D_REF * (ry1 + rz1);
            float u1 = p1 + C_REF * ry1;
            float s1 = (u1 >= 1.0f) ? 1.0f : 0.0f;
            rz1 = D_REF * rz1 + s1;
            qy1 = D_SR * (qy1 + qz1);  qz1 = D_SR * qz1 + s1;
            os1[j] = A_SR * qy1;
        }
        float4 ov0; ov0.x=os0[0]; ov0.y=os0[1]; ov0.z=os0[2]; ov0.w=os0[3];
        float4 ov1; ov1.x=os1[0]; ov1.y=os1[1]; ov1.z=os1[2]; ov1.w=os1[3];
        pout0[t4] = ov0;
        pout1[t4] = ov1;
    }
    float4 z4 = {0.f, 0.f, 0.f, 0.f};
    pout0[125] = z4; pout0[126] = z4; pout0[127] = z4;
    pout1[125] = z4; pout1[126] = z4; pout1[127] = z4;
}

// ---------------------------------------------------------------------------
// fc1 GEMM via V_WMMA_F32_16X16X4_F32 (exact fp32), split-K = 7.
// Grid (t-tile, batch, k-chunk); one wave each; 225 K-steps of 4.
// Partials written to U2p[kc]; deterministic reduction follows.
// ---------------------------------------------------------------------------
__global__ void fc1_wmma_kernel(const float* __restrict__ pack,
                                const float* __restrict__ P1,
                                float* __restrict__ U2p) {
    int nt   = blockIdx.x;        // t-tile 0..31
    int b    = blockIdx.y;        // batch 0..7
    int kc   = blockIdx.z;        // k-chunk 0..6
    int lane = threadIdx.x;       // 0..31
    int n = lane & 15;
    int g = lane >> 4;
    int tbase = nt * 16;
    int kbeg = kc * KCHUNK;

    const v2f*   ap   = (const v2f*)pack;
    const float* p1b  = P1 + (size_t)b * F_IN * TPAD;
    const float* bptr = p1b + (size_t)(2 * g) * TPAD + tbase + n;

    v8f acc0 = {0.f, 0.f, 0.f, 0.f, 0.f, 0.f, 0.f, 0.f};
    v8f acc1 = acc0;

    for (int kk = 0; kk < KCHUNK; ++kk) {
        int k0 = kbeg + kk;
        v2f a0 = ap[(size_t)k0 * 32 + lane];            // w1 rows 0..15
        v2f a1 = ap[(size_t)(K0N + k0) * 32 + lane];    // w1 rows 16..31
        const float* bp = bptr + (size_t)k0 * 4 * TPAD;
        v2f bf;
        bf[0] = bp[0];       // K = 4*k0 + 2g
        bf[1] = bp[TPAD];    // K = 4*k0 + 2g + 1
        acc0 = __builtin_amdgcn_wmma_f32_16x16x4_f32(
                   false, a0, false, bf, (short)0, acc0, false, false);
        acc1 = __builtin_amdgcn_wmma_f32_16x16x4_f32(
                   false, a1, false, bf, (short)0, acc1, false, false);
    }

    float* u2 = U2p + (size_t)kc * U2ELEMS
                    + (size_t)b * HID * TPAD + tbase + n;
#pragma unroll
    for (int r = 0; r < 8; ++r) {
        int m0 = r + 8 * g;                          // rows 0..15
        u2[(size_t)m0 * TPAD]        = acc0[r];
        u2[(size_t)(m0 + 16) * TPAD] = acc1[r];      // rows 16..31
    }
}

// ---------------------------------------------------------------------------
// Deterministic split-K reduction: U2 = sum_kc U2p[kc] (fixed order).
// ---------------------------------------------------------------------------
__global__ void reduce_u2_kernel(const float* __restrict__ U2p,
                                 float* __restrict__ U2) {
    size_t idx = (size_t)blockIdx.x * blockDim.x + threadIdx.x;
    if (idx >= U2ELEMS) return;
    float s = 0.f;
#pragma unroll
    for (int kc = 0; kc < KSPLIT; ++kc)
        s += U2p[(size_t)kc * U2ELEMS + idx];
    U2[idx] = s;
}

// ---------------------------------------------------------------------------
// Stage 2: spike over U2 [B,HID,*], then psp -> P2.
// ---------------------------------------------------------------------------
__global__ void stage2_kernel(const float* __restrict__ U2,
                              float* __restrict__ P2) {
    int idx = blockIdx.x * blockDim.x + threadIdx.x;
    if (idx >= NB * HID) return;
    const float* u  = U2 + (size_t)idx * TPAD;
    float*       p2 = P2 + (size_t)idx * TPAD;
    float rz = 0.f, ry = 0.f, qz = 0.f, qy = 0.f;
    for (int t = 0; t < TT; ++t) {
        ry = D_REF * (ry + rz);
        float uu = u[t] + C_REF * ry;
        float s = (uu >= 1.0f) ? 1.0f : 0.0f;
        rz = D_REF * rz + s;
        qy = D_SR * (qy + qz);  qz = D_SR * qz + s;
        p2[t] = A_SR * qy;
    }
}

// ---------------------------------------------------------------------------
// fc2 dense (fully parallel over b,q,t): U3[b,q,t] = w2[q,:] . P2[b,:,t].
// Lanes map to consecutive t -> coalesced P2 loads; P2/w2 are L2-hot.
// ---------------------------------------------------------------------------
__global__ void fc2_dense_kernel(const float* __restrict__ w2,
                                 const float* __restrict__ P2,
                                 float* __restrict__ U3) {
    int idx = blockIdx.x * blockDim.x + threadIdx.x;
    if (idx >= NB * OUTN * TT) return;
    int t   = idx % TT;
    int bq  = idx / TT;
    int q   = bq % OUTN;
    int b   = bq / OUTN;
    const float* p2b = P2 + (size_t)b * HID * TPAD + t;
    const float* wq  = w2 + q * HID;
    float acc = 0.f;
#pragma unroll
    for (int o = 0; o < HID; ++o)
        acc += wq[o] * p2b[(size_t)o * TPAD];
    U3[idx] = acc;
}

// ---------------------------------------------------------------------------
// Stage 3: final refractory spike over U3 -> out. Only the 3-op recurrence
// remains in the sequential loop (dot product hoisted to fc2_dense).
// ---------------------------------------------------------------------------
__global__ void stage3_kernel(const float* __restrict__ U3,
                              float* __restrict__ out) {
    int idx = blockIdx.x * blockDim.x + threadIdx.x;
    if (idx >= NB * OUTN) return;
    const float* u  = U3  + (size_t)idx * TT;
    float*       ob = out + (size_t)idx * TT;
    float rz = 0.f, ry = 0.f;
    for (int t = 0; t < TT; ++t) {
        ry = D_REF * (ry + rz);
        float uu = u[t] + C_REF * ry;
        float s = (uu >= 1.0f) ? 1.0f : 0.0f;
        rz = D_REF * rz + s;
        ob[t] = s;
    }
}

// ---------------------------------------------------------------------------
extern "C" void kernel_launch(void* const* d_in, const int* in_sizes, int n_in,
                              void* d_out, int out_size, void* d_ws, size_t ws_size,
                              hipStream_t stream) {
    const float* x  = (const float*)d_in[0];   // [8,6300,500]
    const float* w1 = (const float*)d_in[1];   // [32,6300]
    const float* w2 = (const float*)d_in[2];   // [20,32]

    char* ws = (char*)d_ws;
    size_t off = 0;
    float* pack = (float*)(ws + off); off += (size_t)2 * K0N * 64 * sizeof(float);       // 806 KB
    float* U2p  = (float*)(ws + off); off += (size_t)KSPLIT * U2ELEMS * sizeof(float);   // 3.5 MB
    float* U2   = (float*)(ws + off); off += U2ELEMS * sizeof(float);                    // 512 KB
    float* P2   = (float*)(ws + off); off += U2ELEMS * sizeof(float);                    // 512 KB
    float* U3   = (float*)(ws + off); off += (size_t)NB * OUTN * TT * sizeof(float);     // 320 KB
    off = (off + 255) & ~(size_t)255;
    float* P1   = (float*)(ws + off); off += (size_t)NB * F_IN * TPAD * sizeof(float);   // ~103 MB

    repack_w1_kernel<<<(2 * K0N * 64 + 255) / 256, 256, 0, stream>>>(w1, pack);
    stage1_kernel<<<(HSER + 255) / 256, 256, 0, stream>>>(x, P1);
    fc1_wmma_kernel<<<dim3(NTILES, NB, KSPLIT), 32, 0, stream>>>(pack, P1, U2p);
    reduce_u2_kernel<<<(int)((U2ELEMS + 255) / 256), 256, 0, stream>>>(U2p, U2);
    stage2_kernel<<<(NB * HID + 31) / 32, 32, 0, stream>>>(U2, P2);
    fc2_dense_kernel<<<(NB * OUTN * TT + 255) / 256, 256, 0, stream>>>(w2, P2, U3);
    stage3_kernel<<<(NB * OUTN + 31) / 32, 32, 0, stream>>>(U3, (float*)d_out);
}